// Rep_MoVE_47897475285629
// MI455X (gfx1250) — compile-verified
//
#include <hip/hip_runtime.h>
#include <hip/hip_bf16.h>

#define BB   8
#define CC   256
#define HH   64
#define WWD  64
#define EE   9
#define OUTC 256
#define EPSF 1e-5f

typedef __attribute__((ext_vector_type(2))) float v2f;
typedef __attribute__((ext_vector_type(8))) float v8f;

// ---- workspace layout (float offsets) ----
#define WS_WEFF   0        // C*9   = 2304 floats
#define WS_EBIAS  2304     // 256
#define WS_B2     2560     // 256
#define WS_T2     2816     // 256
#define WS_WSWZ   3072     // 64*16*32*2 = 65536 floats (A-fragment swizzled w_out*t2)
// total = 68608 floats = 274,432 bytes

__device__ __forceinline__ float silu(float v) {
    return v / (1.0f + __expf(-v));
}

// Fold E experts + BN + scales into one depthwise 3x3 kernel per channel,
// and fold the output BN into t2/b2.
__global__ __launch_bounds__(256) void fold_kernel(
    const float* __restrict__ w_experts, const float* __restrict__ bn_gamma,
    const float* __restrict__ bn_beta,   const float* __restrict__ bn_mean,
    const float* __restrict__ bn_var,    const float* __restrict__ scales,
    const float* __restrict__ out_gamma, const float* __restrict__ out_beta,
    const float* __restrict__ out_mean,  const float* __restrict__ out_var,
    float* __restrict__ ws)
{
    const int c = threadIdx.x;   // 0..255
    float wacc[9];
#pragma unroll
    for (int k = 0; k < 9; ++k) wacc[k] = 0.0f;
    float bias = 0.0f;
#pragma unroll
    for (int e = 0; e < EE; ++e) {
        const int ec = e * CC + c;
        const float t  = bn_gamma[ec] * rsqrtf(bn_var[ec] + EPSF);
        const float es = t * scales[ec];
        bias += (bn_beta[ec] - bn_mean[ec] * t) * scales[ec];
        const float* wp = w_experts + ec * 9;
#pragma unroll
        for (int k = 0; k < 9; ++k) wacc[k] += es * wp[k];
    }
#pragma unroll
    for (int k = 0; k < 9; ++k) ws[WS_WEFF + c * 9 + k] = wacc[k];
    ws[WS_EBIAS + c] = bias;

    const float t2 = out_gamma[c] * rsqrtf(out_var[c] + EPSF);
    ws[WS_T2 + c] = t2;
    ws[WS_B2 + c] = out_beta[c] - out_mean[c] * t2;
}

// Swizzle w_out * t2 into per-lane A-fragment order:
//   pair index = ((kstep*16 + mtile)*32 + lane), pair = {w[m][k], w[m][k+1]}
//   m = mtile*16 + (lane&15), k = kstep*4 + 2*(lane>>4)
__global__ __launch_bounds__(256) void swz_kernel(
    const float* __restrict__ w_out, float* __restrict__ ws)
{
    const int idx   = blockIdx.x * 256 + threadIdx.x;   // 0..32767
    const int kstep = idx >> 9;
    const int rem   = idx & 511;
    const int mt    = rem >> 5;
    const int lane  = rem & 31;
    const int m     = mt * 16 + (lane & 15);
    const int k     = kstep * 4 + ((lane >> 4) << 1);
    const float t2  = ws[WS_T2 + m];
    ws[WS_WSWZ + idx * 2]     = w_out[m * CC + k]     * t2;
    ws[WS_WSWZ + idx * 2 + 1] = w_out[m * CC + k + 1] * t2;
}

// Fused: depthwise 3x3 conv + bias + SiLU -> LDS (B-fragment order),
// then 256x256 x 64 GEMM via V_WMMA_F32_16X16X4_F32, + b2, SiLU, store.
// One block per (b, h): N-tile = 64 (full row of W), M = all 256 outputs.
__global__ __launch_bounds__(256) void move_kernel(
    const float* __restrict__ x, const float* __restrict__ ws,
    float* __restrict__ out)
{
    __shared__ float smem[CC * WWD];   // 16384 floats = 64 KB, B-fragment swizzled

    const int tid = threadIdx.x;
    const int b   = blockIdx.x >> 6;
    const int h   = blockIdx.x & 63;

    const float* weff  = ws + WS_WEFF;
    const float* ebias = ws + WS_EBIAS;
    const float* b2    = ws + WS_B2;
    const float* wswz  = ws + WS_WSWZ;

    // ---------------- Stage 1: depthwise conv + SiLU into LDS ----------------
    {
        const int w  = tid & 63;    // coalesced along w
        const int cl = tid >> 6;    // 4 channels per sweep
#pragma unroll 2
        for (int ci = 0; ci < 64; ++ci) {
            const int c = (ci << 2) + cl;
            const float* xp = x + (((b * CC + c) * HH + h) * WWD) + w;
            const float* we = weff + c * 9;
            float s = 0.0f;
#pragma unroll
            for (int dy = -1; dy <= 1; ++dy) {
                const int h2 = h + dy;
                if (h2 >= 0 && h2 < HH) {
#pragma unroll
                    for (int dx = -1; dx <= 1; ++dx) {
                        const int w2 = w + dx;
                        if (w2 >= 0 && w2 < WWD)
                            s += xp[dy * WWD + dx] * we[(dy + 1) * 3 + (dx + 1)];
                    }
                }
            }
            s = silu(s + ebias[c]);
            // B-fragment swizzled store: c = kstep*4 + kin, kin = 2*hi + vg
            const int kstep  = c >> 2;
            const int kin    = c & 3;
            const int lane_s = (w & 15) + ((kin >> 1) << 4);
            const int ntile  = w >> 4;
            smem[(((((kstep << 2) + ntile) << 5) + lane_s) << 1) + (kin & 1)] = s;
        }
    }
    __syncthreads();

    // ---------------- Stage 2: GEMM, 8 waves, each 32 M-rows x 64 N-cols ----------------
    const int lane = tid & 31;
    const int wv   = tid >> 5;
    const int mt0  = wv << 1;                 // first of 2 consecutive 16-row M tiles
    const int mo   = (lane >> 4) << 3;        // +8 for hi half of wave

    v8f acc[2][4];
#pragma unroll
    for (int ms = 0; ms < 2; ++ms) {
        const float* bp = b2 + ((mt0 + ms) << 4) + mo;
#pragma unroll
        for (int r = 0; r < 8; ++r) {
            const float bval = bp[r];
#pragma unroll
            for (int ns = 0; ns < 4; ++ns) acc[ms][ns][r] = bval;
        }
    }

    const v2f* aptr = (const v2f*)wswz;
    const v2f* bptr = (const v2f*)smem;
#pragma unroll 4
    for (int kstep = 0; kstep < 64; ++kstep) {
        const v2f a0 = aptr[((kstep << 4) + mt0)     * 32 + lane];
        const v2f a1 = aptr[((kstep << 4) + mt0 + 1) * 32 + lane];
#pragma unroll
        for (int ns = 0; ns < 4; ++ns) {
            const v2f bf = bptr[((kstep << 2) + ns) * 32 + lane];
            acc[0][ns] = __builtin_amdgcn_wmma_f32_16x16x4_f32(
                false, a0, false, bf, (short)0, acc[0][ns], false, false);
            acc[1][ns] = __builtin_amdgcn_wmma_f32_16x16x4_f32(
                false, a1, false, bf, (short)0, acc[1][ns], false, false);
        }
    }

    // ---------------- Epilogue: SiLU + store ----------------
    const int ncol = lane & 15;
#pragma unroll
    for (int ms = 0; ms < 2; ++ms) {
#pragma unroll
        for (int ns = 0; ns < 4; ++ns) {
#pragma unroll
            for (int r = 0; r < 8; ++r) {
                const int o = ((mt0 + ms) << 4) + mo + r;
                const float v = silu(acc[ms][ns][r]);
                out[(((b * OUTC + o) * HH + h) * WWD) + (ns << 4) + ncol] = v;
            }
        }
    }
}

extern "C" void kernel_launch(void* const* d_in, const int* in_sizes, int n_in,
                              void* d_out, int out_size, void* d_ws, size_t ws_size,
                              hipStream_t stream) {
    const float* x         = (const float*)d_in[0];
    const float* w_experts = (const float*)d_in[1];
    const float* bn_gamma  = (const float*)d_in[2];
    const float* bn_beta   = (const float*)d_in[3];
    const float* bn_mean   = (const float*)d_in[4];
    const float* bn_var    = (const float*)d_in[5];
    const float* scales    = (const float*)d_in[6];
    const float* w_out     = (const float*)d_in[7];
    const float* out_gamma = (const float*)d_in[8];
    const float* out_beta  = (const float*)d_in[9];
    const float* out_mean  = (const float*)d_in[10];
    const float* out_var   = (const float*)d_in[11];
    float* out = (float*)d_out;
    float* ws  = (float*)d_ws;

    fold_kernel<<<1, 256, 0, stream>>>(w_experts, bn_gamma, bn_beta, bn_mean,
                                       bn_var, scales, out_gamma, out_beta,
                                       out_mean, out_var, ws);
    swz_kernel<<<128, 256, 0, stream>>>(w_out, ws);
    move_kernel<<<BB * HH, 256, 0, stream>>>(x, ws, out);
}